// PairwisePrediction_54640573939859
// MI455X (gfx1250) — compile-verified
//
#include <hip/hip_runtime.h>

// Problem constants from the reference.
#define BZ 2
#define SZ 512
#define DD 256
#define SLOPE 0.2f

typedef __attribute__((ext_vector_type(2))) float v2f;
typedef __attribute__((ext_vector_type(8))) float v8f;

// One 256-thread block (8 waves) owns a 16-row strip of
// act = leakyrelu(E @ W^T + b). Wave w sweeps N-tiles {2w, 2w+1} with
// V_WMMA_F32_16X16X4_F32 (fp32, exact-precision), fuses bias + LeakyReLU +
// the cw[:D]/cw[D:] dot products into the C-register epilogue, reduces
// across the 16 N-lanes with shuffles, then reduces across the 8 waves
// through LDS in a fixed serial order (deterministic, no atomics).
__global__ __launch_bounds__(256) void proj_scores_kernel(
    const float* __restrict__ E,      // [B*S, D]
    const float* __restrict__ W,      // [D, D]  (proj = E @ W^T)
    const float* __restrict__ bias,   // [D]
    const float* __restrict__ cw,     // [2D]
    float* __restrict__ sL,           // [B*S]
    float* __restrict__ sR)           // [B*S]
{
    __shared__ float redL[8][16];     // [wave][row-in-strip]
    __shared__ float redR[8][16];

    const int tid  = threadIdx.x;
    const int wave = tid >> 5;        // 0..7
    const int lane = tid & 31;
    const int half = lane >> 4;       // 0: lanes 0-15, 1: lanes 16-31
    const int l16  = lane & 15;
    const int m0   = blockIdx.x << 4; // row-tile base

    // A-matrix (16x4 f32) per-lane layout: lanes 0-15 hold K=k,k+1;
    // lanes 16-31 hold K=k+2,k+3 (row M = lane&15). 8-byte vector loads.
    const float* __restrict__ Arow = E + (size_t)(m0 + l16) * DD + (half << 1);

    float accL[8], accR[8];
#pragma unroll
    for (int r = 0; r < 8; ++r) { accL[r] = 0.0f; accR[r] = 0.0f; }

#pragma unroll
    for (int t = 0; t < 2; ++t) {
        const int n0 = ((wave << 1) + t) << 4;   // this wave's N-tile base
        // B-matrix (4x16) mirrors A with N in place of M; B[k][n] = W[n][k].
        const float* __restrict__ Brow =
            W + (size_t)(n0 + l16) * DD + (half << 1);

        v8f c = {};
#pragma unroll 8
        for (int k = 0; k < DD; k += 4) {
            v2f a  = *(const v2f*)(Arow + k);
            v2f bb = *(const v2f*)(Brow + k);
            // 8 args: (neg_a, A, neg_b, B, c_mod, C, reuse_a, reuse_b)
            c = __builtin_amdgcn_wmma_f32_16x16x4_f32(
                false, a, false, bb, (short)0, c, false, false);
        }

        // C/D layout: VGPR r holds M = r + 8*half, N = lane&15.
        const float bn = bias[n0 + l16];
        const float cl = cw[n0 + l16];
        const float cr = cw[DD + n0 + l16];
#pragma unroll
        for (int r = 0; r < 8; ++r) {
            float x = c[r] + bn;
            x = (x >= 0.0f) ? x : SLOPE * x;   // LeakyReLU, exact elementwise
            accL[r] += x * cl;
            accR[r] += x * cr;
        }
    }

    // Reduce across the 16 N-lanes of each half (xor 8,4,2,1 stays in-half),
    // then park this wave's 16 row-partials in LDS.
#pragma unroll
    for (int r = 0; r < 8; ++r) {
        float vl = accL[r], vr = accR[r];
#pragma unroll
        for (int m = 8; m >= 1; m >>= 1) {
            vl += __shfl_xor(vl, m, 32);
            vr += __shfl_xor(vr, m, 32);
        }
        if (l16 == 0) {
            const int row = r + (half << 3);   // 0..15 within strip
            redL[wave][row] = vl;
            redR[wave][row] = vr;
        }
    }

    __syncthreads();

    // Deterministic cross-wave reduction: fixed serial order over wave index.
    if (tid < 32) {
        const int row = tid & 15;
        float s = 0.0f;
        if (tid < 16) {
#pragma unroll
            for (int w = 0; w < 8; ++w) s += redL[w][row];
            sL[m0 + row] = s;
        } else {
#pragma unroll
            for (int w = 0; w < 8; ++w) s += redR[w][row];
            sR[m0 + row] = s;
        }
    }
}

// out[b,i,j] = sR[b*S+i] + sL[b*S+j] + cb. One float4 (4 j's) per thread.
__global__ __launch_bounds__(256) void logits_kernel(
    const float* __restrict__ sL, const float* __restrict__ sR,
    const float* __restrict__ cbp, float* __restrict__ out)
{
    const int idx  = blockIdx.x * blockDim.x + threadIdx.x; // float4 index
    const int jq   = idx & (SZ / 4 - 1);                    // j/4 within row
    const int row  = idx >> 7;                              // b*S + i (SZ/4==128)
    const int bidx = row >> 9;                              // / SZ

    const float base = sR[row] + cbp[0];
    const float4 l = ((const float4*)(sL + (size_t)bidx * SZ))[jq];
    float4 o;
    o.x = base + l.x;
    o.y = base + l.y;
    o.z = base + l.z;
    o.w = base + l.w;
    ((float4*)out)[idx] = o;
}

extern "C" void kernel_launch(void* const* d_in, const int* in_sizes, int n_in,
                              void* d_out, int out_size, void* d_ws, size_t ws_size,
                              hipStream_t stream) {
    (void)in_sizes; (void)n_in; (void)out_size; (void)ws_size;
    const float* E    = (const float*)d_in[0];  // [B,S,D]
    const float* W    = (const float*)d_in[1];  // [D,D]
    const float* bias = (const float*)d_in[2];  // [D]
    const float* cw   = (const float*)d_in[3];  // [2D]
    const float* cb   = (const float*)d_in[4];  // [1]
    float* out = (float*)d_out;

    float* sL = (float*)d_ws;        // [B*S]
    float* sR = sL + BZ * SZ;        // [B*S]

    // 64 blocks x 8 waves; each wave: 2 N-tiles x 64 K-steps of
    // v_wmma_f32_16x16x4_f32, then LDS cross-wave score reduction.
    proj_scores_kernel<<<(BZ * SZ) / 16, 256, 0, stream>>>(E, W, bias, cw, sL, sR);

    // 2*512*512 outputs / 4 per thread / 256 per block = 512 blocks.
    logits_kernel<<<(BZ * SZ * SZ / 4) / 256, 256, 0, stream>>>(sL, sR, cb, out);
}